// LocalContextAttention_47175920779963
// MI455X (gfx1250) — compile-verified
//
#include <hip/hip_runtime.h>
#include <math.h>

// ---------------------------------------------------------------------------
// LocalContextAttention for MI455X (gfx1250, wave32, WMMA)
//   B=4, T=4096, F=H=128, W=9 (band half-width 4), LN eps = 1e-3
// Kernel 1: conv1d(SAME)+relu, q/v/p projections, k = relu(conv)@wk — all via
//           V_WMMA_F32_16X16X4_F32 (fp32 matrix pipe, reference precision).
// Kernel 2: banded softmax-attention + residual + LayerNorm, wave-per-token.
// ---------------------------------------------------------------------------

#define B_   4
#define T_   4096
#define F_   128
#define H_   128
#define W_   9
#define HALO 4          // W_/2

typedef float v2f __attribute__((ext_vector_type(2)));
typedef float v8f __attribute__((ext_vector_type(8)));

static __device__ __forceinline__ v8f wmma4(v2f a, v2f b, v8f c) {
  // (neg_a, A, neg_b, B, c_mod, C, reuse_a, reuse_b)
  return __builtin_amdgcn_wmma_f32_16x16x4_f32(false, a, false, b, (short)0, c,
                                               false, false);
}

// ---------------------------------------------------------------------------
// Kernel 1: per 16-row time tile, 8 waves (one per 16-col output tile).
// VGPR layouts per CDNA5 ISA §7.12.2:
//   A (16x4 f32):  lane 0-15 -> M=lane, K={0,1}; lane 16-31 -> M=lane-16, K={2,3}
//   B (4x16 f32):  lane 0-15 -> N=lane, K={0,1}; lane 16-31 -> N=lane-16, K={2,3}
//   C/D (16x16):   VGPR r -> row r + 8*(lane>=16), col = lane&15
// ---------------------------------------------------------------------------
__global__ __launch_bounds__(256) void lca_proj_kernel(
    const float* __restrict__ x, const float* __restrict__ conv_w,
    const float* __restrict__ conv_b,
    const float* __restrict__ wq, const float* __restrict__ bq,
    const float* __restrict__ wk, const float* __restrict__ bk,
    const float* __restrict__ wv, const float* __restrict__ bv,
    const float* __restrict__ wp, const float* __restrict__ bp,
    float* __restrict__ q, float* __restrict__ k,
    float* __restrict__ v, float* __restrict__ p)
{
  __shared__ float xs[(16 + 2 * HALO) * H_];   // 24 x 128 staged x rows (12 KB)
  __shared__ float lcs[16 * H_];               // relu(conv) tile        (8 KB)

  const int tile = blockIdx.x;                 // 0 .. B*T/16-1
  const int b    = tile / (T_ / 16);
  const int t0   = (tile % (T_ / 16)) * 16;
  const int tid  = threadIdx.x;

  // Cooperative stage of x rows [t0-4, t0+19] with zero padding.
  for (int i = tid; i < 24 * H_; i += 256) {
    const int r  = i >> 7;
    const int c  = i & 127;
    const int tt = t0 - HALO + r;
    xs[i] = (tt >= 0 && tt < T_) ? x[(((size_t)b * T_ + tt) << 7) + c] : 0.0f;
  }
  __syncthreads();

  const int lane  = tid & 31;
  const int wid   = tid >> 5;           // 0..7 -> output column tile
  const int n0    = wid << 4;
  const int mrow  = lane & 15;          // A row index (and C/D column index)
  const int kpair = (lane >> 4) << 1;   // K sub-offset {0 or 2}
  const int ncol  = n0 + mrow;          // B/C column owned by this lane
  const int rbase = (lane >> 4) << 3;   // C/D row base (0 or 8)

  v8f accq, accv, accp, acc;
  {
    const float cq = bq[ncol], cv = bv[ncol], cp = bp[ncol];
#pragma unroll
    for (int r = 0; r < 8; ++r) { accq[r] = cq; accv[r] = cv; accp[r] = cp; }
  }

  // ---- q / v / p : central 16 rows of xs times wq/wv/wp (shared A operand)
  for (int kk = 0; kk < F_; kk += 4) {
    v2f a, bqm, bvm, bpm;
    const float* xa = &xs[(HALO + mrow) * H_ + kk + kpair];
    a.x = xa[0];  a.y = xa[1];
    const int wi = (kk + kpair) * H_ + ncol;
    bqm.x = wq[wi]; bqm.y = wq[wi + H_];
    bvm.x = wv[wi]; bvm.y = wv[wi + H_];
    bpm.x = wp[wi]; bpm.y = wp[wi + H_];
    accq = wmma4(a, bqm, accq);
    accv = wmma4(a, bvm, accv);
    accp = wmma4(a, bpm, accp);
  }
#pragma unroll
  for (int r = 0; r < 8; ++r) {
    const size_t o = (((size_t)b * T_ + t0 + rbase + r) << 7) + ncol;
    q[o] = accq[r];  v[o] = accv[r];  p[o] = accp[r];
  }

  // ---- conv1d: 9 shifted GEMMs accumulated into one tile; relu -> LDS
  {
    const float cb = conv_b[ncol];
#pragma unroll
    for (int r = 0; r < 8; ++r) acc[r] = cb;
  }
  for (int w = 0; w < W_; ++w) {
    const float* cw = conv_w + (size_t)w * F_ * H_;
    for (int kk = 0; kk < F_; kk += 4) {
      v2f a, bm;
      const float* xa = &xs[(w + mrow) * H_ + kk + kpair];
      a.x = xa[0];  a.y = xa[1];
      const int wi = (kk + kpair) * H_ + ncol;
      bm.x = cw[wi]; bm.y = cw[wi + H_];
      acc = wmma4(a, bm, acc);
    }
  }
#pragma unroll
  for (int r = 0; r < 8; ++r)
    lcs[(rbase + r) * H_ + ncol] = fmaxf(acc[r], 0.0f);
  __syncthreads();

  // ---- k = relu(conv) @ wk + bk
  {
    const float cb = bk[ncol];
#pragma unroll
    for (int r = 0; r < 8; ++r) acc[r] = cb;
  }
  for (int kk = 0; kk < F_; kk += 4) {
    v2f a, bm;
    const float* xa = &lcs[mrow * H_ + kk + kpair];
    a.x = xa[0];  a.y = xa[1];
    const int wi = (kk + kpair) * H_ + ncol;
    bm.x = wk[wi]; bm.y = wk[wi + H_];
    acc = wmma4(a, bm, acc);
  }
#pragma unroll
  for (int r = 0; r < 8; ++r)
    k[(((size_t)b * T_ + t0 + rbase + r) << 7) + ncol] = acc[r];
}

// ---------------------------------------------------------------------------
// Kernel 2: wave-per-token banded attention + residual + LayerNorm.
// Lane owns h = 4*lane .. 4*lane+3 (float4 loads). Dot/LN reductions via
// wave32 __shfl_xor.
// ---------------------------------------------------------------------------
static __device__ __forceinline__ float wave_sum(float d) {
#pragma unroll
  for (int off = 16; off > 0; off >>= 1) d += __shfl_xor(d, off, 32);
  return d;
}

__global__ __launch_bounds__(256) void lca_attn_kernel(
    const float* __restrict__ q, const float* __restrict__ k,
    const float* __restrict__ v, const float* __restrict__ p,
    const float* __restrict__ gamma, const float* __restrict__ beta,
    float* __restrict__ out)
{
  const int wid  = threadIdx.x >> 5;
  const int lane = threadIdx.x & 31;
  const int tok  = blockIdx.x * 8 + wid;     // b*T + t
  const int t    = tok & (T_ - 1);
  const int h0   = lane << 2;

  const size_t rowo = ((size_t)tok << 7) + h0;
  const float4 q4 = *(const float4*)(q + rowo);
  const float4 p4 = *(const float4*)(p + rowo);

  float  sc[W_];
  float4 vv[W_];
#pragma unroll
  for (int s = 0; s < W_; ++s) {
    const int st = t - HALO + s;
    if (st >= 0 && st < T_) {                        // uniform across the wave
      const size_t so = (((size_t)tok + (st - t)) << 7) + h0;
      const float4 k4 = *(const float4*)(k + so);
      vv[s] = *(const float4*)(v + so);
      float d = q4.x * k4.x + q4.y * k4.y + q4.z * k4.z + q4.w * k4.w;
      sc[s] = wave_sum(d) * 0.08838834764831845f;    // 1/sqrt(128)
    } else {
      vv[s] = make_float4(0.f, 0.f, 0.f, 0.f);
      sc[s] = -1e9f;                                 // reference band mask
    }
  }

  // softmax over the 9-wide band (numerically stable)
  float m = sc[0];
#pragma unroll
  for (int s = 1; s < W_; ++s) m = fmaxf(m, sc[s]);
  float e[W_], esum = 0.f;
#pragma unroll
  for (int s = 0; s < W_; ++s) { e[s] = __expf(sc[s] - m); esum += e[s]; }
  const float inv = 1.0f / esum;

  float4 o4 = p4;                                    // ctx + x@wp + bp
#pragma unroll
  for (int s = 0; s < W_; ++s) {
    const float a = e[s] * inv;
    o4.x += a * vv[s].x;  o4.y += a * vv[s].y;
    o4.z += a * vv[s].z;  o4.w += a * vv[s].w;
  }

  // LayerNorm over H=128, eps = 1e-3
  const float lsum = wave_sum(o4.x + o4.y + o4.z + o4.w);
  const float lsq  = wave_sum(o4.x * o4.x + o4.y * o4.y +
                              o4.z * o4.z + o4.w * o4.w);
  const float mean = lsum * (1.0f / 128.0f);
  const float var  = lsq * (1.0f / 128.0f) - mean * mean;
  const float rs   = rsqrtf(var + 1e-3f);

  const float4 g4 = *(const float4*)(gamma + h0);
  const float4 b4 = *(const float4*)(beta + h0);
  float4 r;
  r.x = (o4.x - mean) * rs * g4.x + b4.x;
  r.y = (o4.y - mean) * rs * g4.y + b4.y;
  r.z = (o4.z - mean) * rs * g4.z + b4.z;
  r.w = (o4.w - mean) * rs * g4.w + b4.w;
  *(float4*)(out + rowo) = r;
}

// ---------------------------------------------------------------------------
extern "C" void kernel_launch(void* const* d_in, const int* in_sizes, int n_in,
                              void* d_out, int out_size, void* d_ws,
                              size_t ws_size, hipStream_t stream) {
  const float* x      = (const float*)d_in[0];
  const float* conv_w = (const float*)d_in[1];
  const float* conv_b = (const float*)d_in[2];
  const float* wq     = (const float*)d_in[3];
  const float* bq     = (const float*)d_in[4];
  const float* wk     = (const float*)d_in[5];
  const float* bk     = (const float*)d_in[6];
  const float* wv     = (const float*)d_in[7];
  const float* bv     = (const float*)d_in[8];
  const float* wp     = (const float*)d_in[9];
  const float* bp     = (const float*)d_in[10];
  const float* gamma  = (const float*)d_in[11];
  const float* beta   = (const float*)d_in[12];
  float* out = (float*)d_out;

  const size_t n = (size_t)B_ * T_ * H_;       // 2,097,152 floats per buffer
  float* q = (float*)d_ws;                     // 4 buffers -> 32 MB workspace
  float* k = q + n;
  float* v = k + n;
  float* p = v + n;

  lca_proj_kernel<<<(B_ * T_) / 16, 256, 0, stream>>>(
      x, conv_w, conv_b, wq, bq, wk, bk, wv, bv, wp, bp, q, k, v, p);
  lca_attn_kernel<<<(B_ * T_) / 8, 256, 0, stream>>>(
      q, k, v, p, gamma, beta, out);
}